// AtomConv_17532056502701
// MI455X (gfx1250) — compile-verified
//
#include <hip/hip_runtime.h>

typedef __attribute__((ext_vector_type(2))) float v2f;
typedef __attribute__((ext_vector_type(8))) float v8f;

#define NIN 5
#define NOUT 16

// acc = 0, deg = 1.0 (self loop contributes 1 to every node's degree)
__global__ void ac_init_kernel(float* __restrict__ acc, float* __restrict__ deg, int N) {
    int tid = blockIdx.x * blockDim.x + threadIdx.x;
    if (tid < N * NOUT) {
        acc[tid] = 0.0f;
        if ((tid & 15) == 0) deg[tid >> 4] = 1.0f;
    }
}

// deg[col[e]] += 1 for every real edge
__global__ void ac_deg_kernel(const int* __restrict__ ei, float* __restrict__ deg, int E) {
    int e = blockIdx.x * blockDim.x + threadIdx.x;
    if (e < E) atomicAdd(&deg[ei[E + e]], 1.0f);
}

__global__ void ac_dis_kernel(const float* __restrict__ deg, float* __restrict__ dis, int N) {
    int n = blockIdx.x * blockDim.x + threadIdx.x;
    if (n < N) dis[n] = rsqrtf(deg[n]);
}

// x = atom @ W^T + b via V_WMMA_F32_16X16X4_F32.
// K=4 covers features 0..3; feature 4 is folded into the C accumulator.
// One wave handles a 16-node tile (EXEC all-ones within active waves; tile
// predicate is wave-uniform, as WMMA requires).
__global__ void ac_linear_wmma_kernel(const float* __restrict__ atom,
                                      const float* __restrict__ W,
                                      const float* __restrict__ bias,
                                      float* __restrict__ xbuf, int ntiles) {
    int wave = (blockIdx.x * blockDim.x + threadIdx.x) >> 5;
    int lane = threadIdx.x & 31;
    if (wave >= ntiles) return;          // uniform across the wave
    int node = wave * 16;
    int half = lane >> 4;                // 0: K=0,1 / rows 0-7 ; 1: K=2,3 / rows 8-15
    int l    = lane & 15;

    // A fragment (16x4, MxK): lane(half,l) holds atom[node+l][2*half + {0,1}]
    const float* arow = atom + (size_t)(node + l) * NIN;
    v2f a;
    a.x = arow[2 * half + 0];
    a.y = arow[2 * half + 1];

    // B fragment (4x16, KxN): B[k][n] = W[n][k]; lane(half,l) holds W[l][2*half + {0,1}]
    const float* wrow = W + (size_t)l * NIN;
    v2f bm;
    bm.x = wrow[2 * half + 0];
    bm.y = wrow[2 * half + 1];

    // C fragment: C[m][n] = bias[n] + atom[node+m][4] * W[n][4]
    float bn = bias[l];
    float w4 = wrow[4];
    v8f c;
#pragma unroll
    for (int v = 0; v < 8; ++v) {
        int m = v + 8 * half;
        c[v] = bn + atom[(size_t)(node + m) * NIN + 4] * w4;
    }

    // D = A*B + C   ->  v_wmma_f32_16x16x4_f32
    v8f d = __builtin_amdgcn_wmma_f32_16x16x4_f32(
        /*neg_a=*/false, a, /*neg_b=*/false, bm,
        /*c_mod=*/(short)0, c, /*reuse_a=*/false, /*reuse_b=*/false);

    // D layout: VGPR v, lane(half,l) -> row (v + 8*half), col l
#pragma unroll
    for (int v = 0; v < 8; ++v) {
        int m = v + 8 * half;
        xbuf[(size_t)(node + m) * NOUT + l] = d[v];
    }
}

// Scalar fallback for a node-count tail (N % 16 != 0); unused for N = 100000.
__global__ void ac_linear_tail_kernel(const float* __restrict__ atom,
                                      const float* __restrict__ W,
                                      const float* __restrict__ bias,
                                      float* __restrict__ xbuf, int start, int N) {
    int tid = blockIdx.x * blockDim.x + threadIdx.x;
    int n = start + (tid >> 4);
    if (n >= N) return;
    int j = tid & 15;
    float s = bias[j];
#pragma unroll
    for (int k = 0; k < NIN; ++k) s += atom[(size_t)n * NIN + k] * W[j * NIN + k];
    xbuf[(size_t)n * NOUT + j] = s;
}

// One (edge, feature) pair per thread: coalesced 16-wide gather of x[row],
// f32 atomic scatter into L2-resident acc[col].
__global__ void ac_scatter_kernel(const int* __restrict__ ei,
                                  const float* __restrict__ xbuf,
                                  const float* __restrict__ dis,
                                  float* __restrict__ acc, int E) {
    long long tid = (long long)blockIdx.x * blockDim.x + threadIdx.x;
    int e = (int)(tid >> 4);
    if (e >= E) return;
    int j = (int)(tid & 15);
    int r = ei[e];
    int c = ei[E + e];
    float norm = dis[r] * dis[c];
    atomicAdd(&acc[(size_t)c * NOUT + j], norm * xbuf[(size_t)r * NOUT + j]);
}

// out = relu(acc + (1/deg[n]) * x[n])  ; self-loop norm = dis[n]^2
__global__ void ac_finish_kernel(const float* __restrict__ acc,
                                 const float* __restrict__ xbuf,
                                 const float* __restrict__ dis,
                                 float* __restrict__ out, int N) {
    int tid = blockIdx.x * blockDim.x + threadIdx.x;
    if (tid >= N * NOUT) return;
    int n = tid >> 4;
    float selfn = dis[n] * dis[n];
    float v = acc[tid] + selfn * xbuf[tid];
    out[tid] = v > 0.0f ? v : 0.0f;
}

extern "C" void kernel_launch(void* const* d_in, const int* in_sizes, int n_in,
                              void* d_out, int out_size, void* d_ws, size_t ws_size,
                              hipStream_t stream) {
    const float* atom = (const float*)d_in[0];
    const int*   ei   = (const int*)d_in[1];   // JAX default x64-off -> int32
    const float* W    = (const float*)d_in[2];
    const float* bias = (const float*)d_in[3];
    float* out = (float*)d_out;

    int N = in_sizes[0] / NIN;
    int E = in_sizes[1] / 2;

    float* xbuf = (float*)d_ws;                       // N*16 f32
    float* acc  = xbuf + (size_t)N * NOUT;            // N*16 f32
    float* deg  = acc + (size_t)N * NOUT;             // N f32
    float* dis  = deg + N;                            // N f32

    const int T = 256;
    ac_init_kernel<<<(N * NOUT + T - 1) / T, T, 0, stream>>>(acc, deg, N);
    ac_deg_kernel<<<(E + T - 1) / T, T, 0, stream>>>(ei, deg, E);

    int ntiles = N / 16;
    if (ntiles > 0)
        ac_linear_wmma_kernel<<<(ntiles + 7) / 8, 256, 0, stream>>>(atom, W, bias, xbuf, ntiles);
    int rem = N - ntiles * 16;
    if (rem > 0)
        ac_linear_tail_kernel<<<(rem * NOUT + T - 1) / T, T, 0, stream>>>(atom, W, bias, xbuf,
                                                                          ntiles * 16, N);

    ac_dis_kernel<<<(N + T - 1) / T, T, 0, stream>>>(deg, dis, N);

    long long sthreads = (long long)E * NOUT;
    ac_scatter_kernel<<<(unsigned)((sthreads + T - 1) / T), T, 0, stream>>>(ei, xbuf, dis, acc, E);

    ac_finish_kernel<<<(N * NOUT + T - 1) / T, T, 0, stream>>>(acc, xbuf, dis, out, N);
}